// GATNet_8564164788982
// MI455X (gfx1250) — compile-verified
//
#include <hip/hip_runtime.h>
#include <hip/hip_bf16.h>
#include <math.h>

// ---------------------------------------------------------------------------
// GATNet on gfx1250 (MI455X): fp32 WMMA (16x16x4) for the GEMMs, atomic-based
// segment max / softmax / aggregation for the edge stages. All scratch in d_ws.
// ---------------------------------------------------------------------------

typedef __attribute__((ext_vector_type(2))) float v2f;
typedef __attribute__((ext_vector_type(8))) float v8f;

#define NEG_SLOPE 0.2f

// ------------------------------- utility -----------------------------------

__global__ void fill_f32(float* __restrict__ p, long n, float v) {
    long i = (long)blockIdx.x * blockDim.x + threadIdx.x;
    if (i < n) p[i] = v;
}

__device__ __forceinline__ void atomicMaxFloat(float* addr, float val) {
    // standard two-op trick: non-negative floats order like ints,
    // negative floats order inversely like uints.
    if (val >= 0.0f) {
        atomicMax((int*)addr, __float_as_int(val));
    } else {
        atomicMin((unsigned int*)addr, __float_as_uint(val));
    }
}

__device__ __forceinline__ void edge_sd(const int* __restrict__ ei, int e, int E0,
                                        int* s, int* d) {
    if (e < E0) { *s = ei[e]; *d = ei[E0 + e]; }
    else        { *s = e - E0; *d = e - E0; }       // self loops appended
}

__device__ __forceinline__ float leaky(float v) {
    return v > 0.0f ? v : NEG_SLOPE * v;
}

// ------------------------- fp32 WMMA GEMM (16x16x4) -------------------------
// C[M,N] = A[M,K] @ B[K,N].  M,N multiples of 16; K multiple of 4.
// One wave32 per 16x16 output tile.  A: lanes 0-15 carry K={k,k+1},
// lanes 16-31 carry K={k+2,k+3}.  B mirrored with N across lanes.
__global__ void gemm_wmma_f32(const float* __restrict__ A,
                              const float* __restrict__ B,
                              float* __restrict__ C,
                              int M, int K, int N) {
    int wave = (blockIdx.x * blockDim.x + threadIdx.x) >> 5;
    int lane = threadIdx.x & 31;
    int tilesN = N >> 4;
    int tiles  = (M >> 4) * tilesN;
    if (wave >= tiles) return;                      // wave-uniform: EXEC all-1s inside
    int tm = wave / tilesN, tn = wave % tilesN;
    int row0 = tm << 4, col0 = tn << 4;
    int mr   = lane & 15;                           // M (for A) / N (for B) index
    int koff = (lane >> 4) << 1;                    // lanes 16-31 take K+2,K+3

    v8f c = {};
    for (int k = 0; k < K; k += 4) {
        v2f a, b;
        a.x = A[(size_t)(row0 + mr) * K + k + koff];
        a.y = A[(size_t)(row0 + mr) * K + k + koff + 1];
        b.x = B[(size_t)(k + koff)     * N + col0 + mr];
        b.y = B[(size_t)(k + koff + 1) * N + col0 + mr];
        c = __builtin_amdgcn_wmma_f32_16x16x4_f32(
                /*neg_a=*/false, a, /*neg_b=*/false, b,
                /*c_mod=*/(short)0, c, /*reuse_a=*/false, /*reuse_b=*/false);
    }
    // C/D layout: vgpr i, lanes 0-15 -> row i, lanes 16-31 -> row i+8; col = lane&15
    int rbase = (lane >> 4) << 3;
#pragma unroll
    for (int i = 0; i < 8; ++i)
        C[(size_t)(row0 + rbase + i) * N + col0 + mr] = c[i];
}

// -------------------- per-node attention logits (2 dots) --------------------
__global__ void node_dot2(const float* __restrict__ h,
                          const float* __restrict__ asrc,
                          const float* __restrict__ adst,
                          float* __restrict__ os, float* __restrict__ od,
                          int n, int F) {
    int wave = (blockIdx.x * blockDim.x + threadIdx.x) >> 5;
    int lane = threadIdx.x & 31;
    if (wave >= n) return;
    const float* hr = h + (size_t)wave * F;
    float s = 0.0f, d = 0.0f;
    for (int f = lane; f < F; f += 32) {
        float v = hr[f];
        s += v * asrc[f];
        d += v * adst[f];
    }
    for (int off = 16; off; off >>= 1) {
        s += __shfl_down(s, off, 32);
        d += __shfl_down(d, off, 32);
    }
    if (lane == 0) { os[wave] = s; od[wave] = d; }
}

// ------------------------------ edge stages --------------------------------
__global__ void edge_max(const int* __restrict__ ei, int E0, int Etot,
                         const float* __restrict__ as_, const float* __restrict__ ad_,
                         float* __restrict__ emax) {
    int e = blockIdx.x * blockDim.x + threadIdx.x;
    if (e >= Etot) return;
    int s, d; edge_sd(ei, e, E0, &s, &d);
    atomicMaxFloat(&emax[d], leaky(as_[s] + ad_[d]));
}

__global__ void edge_expsum(const int* __restrict__ ei, int E0, int Etot,
                            const float* __restrict__ as_, const float* __restrict__ ad_,
                            const float* __restrict__ emax, float* __restrict__ denom) {
    int e = blockIdx.x * blockDim.x + threadIdx.x;
    if (e >= Etot) return;
    int s, d; edge_sd(ei, e, E0, &s, &d);
    float v = leaky(as_[s] + ad_[d]);
    atomicAdd(&denom[d], expf(v - emax[d]));
}

// wave per edge: out[dst,:] += alpha * h[src,:]
__global__ void edge_aggregate(const int* __restrict__ ei, int E0, int Etot,
                               const float* __restrict__ as_, const float* __restrict__ ad_,
                               const float* __restrict__ emax, const float* __restrict__ denom,
                               const float* __restrict__ h, float* __restrict__ out, int F) {
    int wave = (blockIdx.x * blockDim.x + threadIdx.x) >> 5;
    int lane = threadIdx.x & 31;
    if (wave >= Etot) return;
    int s, d; edge_sd(ei, wave, E0, &s, &d);
    float v = leaky(as_[s] + ad_[d]);
    float w = expf(v - emax[d]) / denom[d];
    const float* hs = h + (size_t)s * F;
    float* od = out + (size_t)d * F;
    for (int f = lane; f < F; f += 32)
        atomicAdd(&od[f], w * hs[f]);
}

__global__ void bias_relu(float* __restrict__ x, const float* __restrict__ b,
                          long n, int fmask) {
    long i = (long)blockIdx.x * blockDim.x + threadIdx.x;
    if (i >= n) return;
    float v = x[i] + b[i & fmask];
    x[i] = v > 0.0f ? v : 0.0f;
}

// ----------------------------- pooling + FC --------------------------------
__global__ void pool_sum(const float* __restrict__ h, const int* __restrict__ batch,
                         float* __restrict__ sums, float* __restrict__ cnts, int n) {
    int wave = (blockIdx.x * blockDim.x + threadIdx.x) >> 5;
    int lane = threadIdx.x & 31;
    if (wave >= n) return;
    int g = batch[wave];
    const float* hr = h + (size_t)wave * 64;
    for (int f = lane; f < 64; f += 32)
        atomicAdd(&sums[g * 64 + f], hr[f]);
    if (lane == 0) atomicAdd(&cnts[g], 1.0f);
}

__global__ void pool_fc(const float* __restrict__ sums, const float* __restrict__ cnts,
                        const float* __restrict__ fcW, const float* __restrict__ fcb,
                        float* __restrict__ out, int G) {
    int g = blockIdx.x * blockDim.x + threadIdx.x;
    if (g >= G) return;
    float c = cnts[g]; c = c > 1.0f ? c : 1.0f;
    float o0 = fcb[0], o1 = fcb[1];
    for (int k = 0; k < 64; ++k) {
        float p = sums[g * 64 + k] / c;
        o0 += p * fcW[k * 2 + 0];
        o1 += p * fcW[k * 2 + 1];
    }
    out[g * 2 + 0] = o0;
    out[g * 2 + 1] = o1;
}

// ------------------------------- launcher ----------------------------------

extern "C" void kernel_launch(void* const* d_in, const int* in_sizes, int n_in,
                              void* d_out, int out_size, void* d_ws, size_t ws_size,
                              hipStream_t stream) {
    (void)n_in; (void)out_size; (void)ws_size;

    const float* x      = (const float*)d_in[0];
    const int*   ei     = (const int*)  d_in[1];
    const int*   batch  = (const int*)  d_in[2];
    const float* W1     = (const float*)d_in[3];
    const float* a_src1 = (const float*)d_in[4];
    const float* a_dst1 = (const float*)d_in[5];
    const float* b1     = (const float*)d_in[6];
    const float* W2     = (const float*)d_in[7];
    const float* a_src2 = (const float*)d_in[8];
    const float* a_dst2 = (const float*)d_in[9];
    const float* b2     = (const float*)d_in[10];
    const float* fcW    = (const float*)d_in[11];
    const float* fcb    = (const float*)d_in[12];
    float* out = (float*)d_out;

    const int N    = in_sizes[0] / 128;   // 10000 (multiple of 16)
    const int E0   = in_sizes[1] / 2;     // 160000
    const int Etot = E0 + N;              // +self loops
    const int G    = 64;

    // workspace carve-out (floats)
    float* ws = (float*)d_ws;
    size_t o = 0;
    float* h1    = ws + o; o += (size_t)N * 512;
    float* out1  = ws + o; o += (size_t)N * 512;
    float* h2    = ws + o; o += (size_t)N * 64;
    float* out2  = ws + o; o += (size_t)N * 64;
    float* as_   = ws + o; o += N;
    float* ad_   = ws + o; o += N;
    float* emax  = ws + o; o += N;
    float* denom = ws + o; o += N;
    float* sums  = ws + o; o += (size_t)G * 64;
    float* cnts  = ws + o; o += G;

    const int TB = 256;
    auto cdiv = [](long a, long b) { return (int)((a + b - 1) / b); };

    const float NEG_INF = -__builtin_huge_valf();

    // ---------------- layer 1 (F = 512) ----------------
    {
        int tiles = (N / 16) * (512 / 16);
        gemm_wmma_f32<<<cdiv((long)tiles * 32, TB), TB, 0, stream>>>(x, W1, h1, N, 128, 512);

        node_dot2<<<cdiv((long)N * 32, TB), TB, 0, stream>>>(h1, a_src1, a_dst1, as_, ad_, N, 512);

        fill_f32<<<cdiv(N, TB), TB, 0, stream>>>(emax, N, NEG_INF);
        fill_f32<<<cdiv(N, TB), TB, 0, stream>>>(denom, N, 0.0f);
        fill_f32<<<cdiv((long)N * 512, TB), TB, 0, stream>>>(out1, (long)N * 512, 0.0f);

        edge_max   <<<cdiv(Etot, TB), TB, 0, stream>>>(ei, E0, Etot, as_, ad_, emax);
        edge_expsum<<<cdiv(Etot, TB), TB, 0, stream>>>(ei, E0, Etot, as_, ad_, emax, denom);
        edge_aggregate<<<cdiv((long)Etot * 32, TB), TB, 0, stream>>>(
            ei, E0, Etot, as_, ad_, emax, denom, h1, out1, 512);

        bias_relu<<<cdiv((long)N * 512, TB), TB, 0, stream>>>(out1, b1, (long)N * 512, 511);
    }

    // ---------------- layer 2 (F = 64) ----------------
    {
        int tiles = (N / 16) * (64 / 16);
        gemm_wmma_f32<<<cdiv((long)tiles * 32, TB), TB, 0, stream>>>(out1, W2, h2, N, 512, 64);

        node_dot2<<<cdiv((long)N * 32, TB), TB, 0, stream>>>(h2, a_src2, a_dst2, as_, ad_, N, 64);

        fill_f32<<<cdiv(N, TB), TB, 0, stream>>>(emax, N, NEG_INF);
        fill_f32<<<cdiv(N, TB), TB, 0, stream>>>(denom, N, 0.0f);
        fill_f32<<<cdiv((long)N * 64, TB), TB, 0, stream>>>(out2, (long)N * 64, 0.0f);

        edge_max   <<<cdiv(Etot, TB), TB, 0, stream>>>(ei, E0, Etot, as_, ad_, emax);
        edge_expsum<<<cdiv(Etot, TB), TB, 0, stream>>>(ei, E0, Etot, as_, ad_, emax, denom);
        edge_aggregate<<<cdiv((long)Etot * 32, TB), TB, 0, stream>>>(
            ei, E0, Etot, as_, ad_, emax, denom, h2, out2, 64);

        bias_relu<<<cdiv((long)N * 64, TB), TB, 0, stream>>>(out2, b2, (long)N * 64, 63);
    }

    // ---------------- pool + FC ----------------
    fill_f32<<<cdiv((long)G * 64 + G, TB), TB, 0, stream>>>(sums, (long)G * 64 + G, 0.0f); // sums+cnts contiguous
    pool_sum<<<cdiv((long)N * 32, TB), TB, 0, stream>>>(out2, batch, sums, cnts, N);
    pool_fc<<<1, G, 0, stream>>>(sums, cnts, fcW, fcb, out, G);
}